// GATEncoder_67336497266936
// MI455X (gfx1250) — compile-verified
//
#include <hip/hip_runtime.h>

// Problem constants (match reference)
#define NN    50000
#define EE    500000
#define INF_  512
#define HH    4
#define CC    64
#define OUTF  256          // H*C
#define NEG_SLOPE 0.2f

#define ROWS  128          // GEMM block row tile
#define KC    16           // GEMM K chunk (LDS, double buffered)

typedef __attribute__((ext_vector_type(2))) float v2f;
typedef __attribute__((ext_vector_type(8))) float v8f;
typedef __attribute__((ext_vector_type(4))) int   v4i;

// ---------------------------------------------------------------------------
// CDNA5 async global->LDS copies (ASYNCcnt path), with safe fallback.
// Builtin signature (from hipcc diagnostic): params are pointers to v4i.
// ---------------------------------------------------------------------------
#if defined(__has_builtin)
# if __has_builtin(__builtin_amdgcn_global_load_async_to_lds_b128) && \
     __has_builtin(__builtin_amdgcn_s_wait_asynccnt)
#  define USE_ASYNC_LDS 1
# endif
#endif

typedef __attribute__((address_space(1))) v4i* gv4i_p;   // global v4i*
typedef __attribute__((address_space(3))) v4i* lv4i_p;   // LDS v4i*

__device__ __forceinline__ void cp_async16(const float* g, float* l) {
#ifdef USE_ASYNC_LDS
  // Flat low 32 bits == workgroup-relative LDS offset (ISA aperture truncation).
  __builtin_amdgcn_global_load_async_to_lds_b128(
      (gv4i_p)(uintptr_t)g, (lv4i_p)(unsigned)(uintptr_t)l, 0, 0);
#else
  *(float4*)l = *(const float4*)g;
#endif
}
__device__ __forceinline__ void cp_async_wait() {
#ifdef USE_ASYNC_LDS
  __builtin_amdgcn_s_wait_asynccnt(0);
#endif
}

// ---- ordered-uint encoding so atomicMax(uint) == float max ----
__device__ __forceinline__ unsigned f2ord(float f) {
  unsigned u = __float_as_uint(f);
  return (u & 0x80000000u) ? ~u : (u | 0x80000000u);
}
__device__ __forceinline__ float ord2f(unsigned u) {
  return (u & 0x80000000u) ? __uint_as_float(u & 0x7fffffffu) : __uint_as_float(~u);
}

__device__ __forceinline__ void edge_sd(const int* __restrict__ ei, int e, int& s, int& d) {
  if (e < EE) { s = ei[e]; d = ei[EE + e]; }       // edge_index[0][e], edge_index[1][e]
  else        { s = d = e - EE; }                   // self loops
}

// ---------------------------------------------------------------------------
// Kernel 0: zero the accumulation buffer (d_out) + denom, init max buffer
// ---------------------------------------------------------------------------
__global__ void gat_init(float* __restrict__ out, float* __restrict__ denom,
                         unsigned* __restrict__ mbuf) {
  int t = blockIdx.x * blockDim.x + threadIdx.x;
  if (t < NN * OUTF) out[t] = 0.0f;
  if (t < NN * HH) { denom[t] = 0.0f; mbuf[t] = 0u; }  // 0u = ordered -inf key
}

// ---------------------------------------------------------------------------
// Kernel 1: h = x @ W via V_WMMA_F32_16X16X4_F32.
// Block = 256 threads (8 waves). Block tile: 128 rows x 256 cols.
// Wave grid 2x4: each wave computes a 64x64 tile (4x4 WMMA tiles, acc = 16*v8f).
// Double-buffered LDS staging through the async-to-LDS path (ASYNCcnt).
// ---------------------------------------------------------------------------
__global__ __launch_bounds__(256) void gat_gemm(const float* __restrict__ x,
                                                const float* __restrict__ W,
                                                float* __restrict__ h) {
  __shared__ float xs[2][ROWS][KC + 4];     // 2 x 128 x 20 floats (20 KB)
  __shared__ float ws[2][KC][OUTF + 4];     // 2 x  16 x 260 floats (33 KB)

  const int tid  = threadIdx.x;
  const int wave = tid >> 5;
  const int lane = tid & 31;
  const int half = lane >> 4;    // 0: lanes 0-15, 1: lanes 16-31
  const int l    = lane & 15;
  const int rg   = wave >> 2;    // row group (64 rows)
  const int cg   = wave & 3;     // col group (64 cols)
  const int rowBase = blockIdx.x * ROWS;

  v8f acc[4][4] = {};

  auto stage = [&](int buf, int chunk) {
    const int k0 = chunk * KC;
    #pragma unroll
    for (int i = 0; i < 2; ++i) {             // xs: 128x16 floats = 512 float4
      int idx = tid + i * 256;
      int r = idx >> 2, c4 = (idx & 3) * 4;
      int grow = rowBase + r;
      if (grow < NN)                           // OOB A-rows only feed OOB D-rows
        cp_async16(x + (size_t)grow * INF_ + k0 + c4, &xs[buf][r][c4]);
    }
    #pragma unroll
    for (int i = 0; i < 4; ++i) {             // ws: 16x256 floats = 1024 float4
      int idx = tid + i * 256;
      int r = idx >> 6, c4 = (idx & 63) * 4;
      cp_async16(W + (size_t)(k0 + r) * OUTF + c4, &ws[buf][r][c4]);
    }
  };

  stage(0, 0);
  cp_async_wait();
  __syncthreads();

  const int NCH = INF_ / KC;                  // 32 chunks
  for (int c = 0; c < NCH; ++c) {
    const int cur = c & 1;
    if (c + 1 < NCH) stage(cur ^ 1, c + 1);   // prefetch next chunk (other buffer)

    #pragma unroll
    for (int kk = 0; kk < KC; kk += 4) {
      // A frag (16x4 f32): lanes 0-15 hold K=kk,kk+1; lanes 16-31 hold K=kk+2,kk+3
      v2f a[4], b[4];
      #pragma unroll
      for (int rt = 0; rt < 4; ++rt) {
        a[rt].x = xs[cur][rg * 64 + rt * 16 + l][kk + 2 * half + 0];
        a[rt].y = xs[cur][rg * 64 + rt * 16 + l][kk + 2 * half + 1];
      }
      #pragma unroll
      for (int ct = 0; ct < 4; ++ct) {
        b[ct].x = ws[cur][kk + 2 * half + 0][cg * 64 + ct * 16 + l];
        b[ct].y = ws[cur][kk + 2 * half + 1][cg * 64 + ct * 16 + l];
      }
      #pragma unroll
      for (int rt = 0; rt < 4; ++rt)
        #pragma unroll
        for (int ct = 0; ct < 4; ++ct)
          acc[rt][ct] = __builtin_amdgcn_wmma_f32_16x16x4_f32(
              false, a[rt], false, b[ct], (short)0, acc[rt][ct], false, false);
    }

    cp_async_wait();      // our prefetch for c+1 done (per-wave ASYNCcnt)
    __syncthreads();      // everyone done reading buf `cur` + all slices landed
  }

  // C/D layout: VGPR r holds M=r (lanes 0-15) / M=8+r (lanes 16-31), N=lane&15
  #pragma unroll
  for (int rt = 0; rt < 4; ++rt)
    #pragma unroll
    for (int ct = 0; ct < 4; ++ct)
      #pragma unroll
      for (int r = 0; r < 8; ++r) {
        int grow = rowBase + rg * 64 + rt * 16 + r + 8 * half;
        int gcol = cg * 64 + ct * 16 + l;
        if (grow < NN) h[(size_t)grow * OUTF + gcol] = acc[rt][ct][r];
      }
}

// ---------------------------------------------------------------------------
// Kernel 2: a_src[n,h] = <h[n,h,:], att_src[h,:]> ; same for a_dst
// ---------------------------------------------------------------------------
__global__ void gat_att(const float* __restrict__ h, const float* __restrict__ att_src,
                        const float* __restrict__ att_dst,
                        float* __restrict__ a_src, float* __restrict__ a_dst) {
  int t = blockIdx.x * blockDim.x + threadIdx.x;
  if (t >= NN * HH) return;
  int n = t / HH, hd = t % HH;
  const float* hp = h + (size_t)n * OUTF + hd * CC;
  const float* as = att_src + hd * CC;
  const float* ad = att_dst + hd * CC;
  float s = 0.f, d = 0.f;
  #pragma unroll 8
  for (int c = 0; c < CC; ++c) { float v = hp[c]; s += v * as[c]; d += v * ad[c]; }
  a_src[t] = s; a_dst[t] = d;
}

// ---------------------------------------------------------------------------
// Kernel 3: per-target segment max of leaky_relu(a_src[src]+a_dst[dst])
// ---------------------------------------------------------------------------
__global__ void gat_max(const int* __restrict__ ei, const float* __restrict__ a_src,
                        const float* __restrict__ a_dst, unsigned* __restrict__ mbuf) {
  int t = blockIdx.x * blockDim.x + threadIdx.x;
  if (t >= (EE + NN) * HH) return;
  int e = t >> 2, hd = t & 3;
  int s, d; edge_sd(ei, e, s, d);
  float v = a_src[s * HH + hd] + a_dst[d * HH + hd];
  v = (v >= 0.f) ? v : NEG_SLOPE * v;
  atomicMax(&mbuf[d * HH + hd], f2ord(v));
}

// ---------------------------------------------------------------------------
// Kernel 4: eexp = exp(e - m[dst]); denom[dst] += eexp
// ---------------------------------------------------------------------------
__global__ void gat_expsum(const int* __restrict__ ei, const float* __restrict__ a_src,
                           const float* __restrict__ a_dst, const unsigned* __restrict__ mbuf,
                           float* __restrict__ denom, float* __restrict__ eexp) {
  int t = blockIdx.x * blockDim.x + threadIdx.x;
  if (t >= (EE + NN) * HH) return;
  int e = t >> 2, hd = t & 3;
  int s, d; edge_sd(ei, e, s, d);
  float v = a_src[s * HH + hd] + a_dst[d * HH + hd];
  v = (v >= 0.f) ? v : NEG_SLOPE * v;
  float ex = expf(v - ord2f(mbuf[d * HH + hd]));
  eexp[t] = ex;
  atomicAdd(&denom[d * HH + hd], ex);
}

// ---------------------------------------------------------------------------
// Kernel 5: out[dst, c] += alpha * h[src, c]   (one block per edge, 256 lanes)
// ---------------------------------------------------------------------------
__global__ __launch_bounds__(256) void gat_scatter(const int* __restrict__ ei,
                                                   const float* __restrict__ h,
                                                   const float* __restrict__ eexp,
                                                   const float* __restrict__ denom,
                                                   float* __restrict__ out) {
  int e = blockIdx.x;
  int c = threadIdx.x;           // 0..255
  int hd = c >> 6;               // head = c / 64
  int s, d; edge_sd(ei, e, s, d);
  float alpha = eexp[e * HH + hd] / (denom[d * HH + hd] + 1e-16f);
  atomicAdd(&out[(size_t)d * OUTF + c], alpha * h[(size_t)s * OUTF + c]);
}

// ---------------------------------------------------------------------------
// Kernel 6: out = PReLU(out + bias)
// ---------------------------------------------------------------------------
__global__ void gat_final(float* __restrict__ out, const float* __restrict__ bias,
                          const float* __restrict__ pa) {
  int t = blockIdx.x * blockDim.x + threadIdx.x;
  if (t >= NN * OUTF) return;
  float v = out[t] + bias[t & (OUTF - 1)];
  out[t] = (v >= 0.f) ? v : pa[0] * v;
}

// ---------------------------------------------------------------------------
extern "C" void kernel_launch(void* const* d_in, const int* in_sizes, int n_in,
                              void* d_out, int out_size, void* d_ws, size_t ws_size,
                              hipStream_t stream) {
  const float* x       = (const float*)d_in[0];
  const int*   ei      = (const int*)  d_in[1];   // [2, E]
  const float* W       = (const float*)d_in[2];
  const float* att_src = (const float*)d_in[3];
  const float* att_dst = (const float*)d_in[4];
  const float* bias    = (const float*)d_in[5];
  const float* pa      = (const float*)d_in[6];
  float* out = (float*)d_out;

  // workspace layout (~63 MB)
  float*    h      = (float*)d_ws;                           // N*256
  float*    a_src  = h + (size_t)NN * OUTF;                  // N*H
  float*    a_dst  = a_src + (size_t)NN * HH;                // N*H
  unsigned* mbuf   = (unsigned*)(a_dst + (size_t)NN * HH);   // N*H
  float*    denom  = (float*)(mbuf + (size_t)NN * HH);       // N*H
  float*    eexp   = denom + (size_t)NN * HH;                // (E+N)*H

  {
    int total = NN * OUTF;
    gat_init<<<(total + 255) / 256, 256, 0, stream>>>(out, denom, mbuf);
  }
  gat_gemm<<<(NN + ROWS - 1) / ROWS, 256, 0, stream>>>(x, W, h);
  {
    int total = NN * HH;
    gat_att<<<(total + 255) / 256, 256, 0, stream>>>(h, att_src, att_dst, a_src, a_dst);
  }
  {
    int total = (EE + NN) * HH;
    gat_max<<<(total + 255) / 256, 256, 0, stream>>>(ei, a_src, a_dst, mbuf);
    gat_expsum<<<(total + 255) / 256, 256, 0, stream>>>(ei, a_src, a_dst, mbuf, denom, eexp);
  }
  gat_scatter<<<EE + NN, 256, 0, stream>>>(ei, h, eexp, denom, out);
  {
    int total = NN * OUTF;
    gat_final<<<(total + 255) / 256, 256, 0, stream>>>(out, bias, pa);
  }
}